// LocalChargeEnergy_549755813995
// MI455X (gfx1250) — compile-verified
//
#include <hip/hip_runtime.h>

typedef float v2f __attribute__((ext_vector_type(2)));
typedef float v8f __attribute__((ext_vector_type(8)));

#define NF 128

__global__ void zero_f32_kernel(float* __restrict__ p, int n) {
    int i = blockIdx.x * blockDim.x + threadIdx.x;
    if (i < n) p[i] = 0.0f;
}

// Accumulate a 128-length dot-product contribution for 16 atoms into the
// wave-wide 16x16 f32 WMMA accumulator. F points at this lane's row slice
// (row*NF + coff), W at the weight vector (+ coff).
__device__ __forceinline__ void accum_term(const float* __restrict__ F,
                                           const float* __restrict__ W,
                                           v8f& c) {
#pragma unroll
    for (int k0 = 0; k0 < NF; k0 += 4) {
        v2f a = *(const v2f*)(F + k0);   // A: 16x4 f32 tile slice (8B/lane)
        v2f b = *(const v2f*)(W + k0);   // B: 4x16 broadcast of weight chunk
        c = __builtin_amdgcn_wmma_f32_16x16x4_f32(
                /*neg_a=*/false, a, /*neg_b=*/false, b,
                /*c_mod=*/(short)0, c, /*reuse_a=*/false, /*reuse_b=*/false);
    }
}

__global__ __launch_bounds__(256)
void charge_energy_kernel(const float* __restrict__ charges,
                          const float* __restrict__ f0,
                          const float* __restrict__ f1,
                          const float* __restrict__ f2,
                          const int*   __restrict__ mol_index,
                          const float* __restrict__ w0,
                          const float* __restrict__ w1,
                          const float* __restrict__ w2,
                          const float* __restrict__ b1p,
                          const float* __restrict__ b2p,
                          float* __restrict__ mol_out,
                          float* __restrict__ atom_out,
                          int n_atoms) {
    const int lane = threadIdx.x & 31;          // wave32
    const int waveInBlock = threadIdx.x >> 5;
    const int wave = blockIdx.x * (blockDim.x >> 5) + waveInBlock;
    const int base = wave * 16;                 // 16 atoms per wave
    if (base >= n_atoms) return;                // uniform per wave: EXEC stays all-1s

    // A-matrix 16x4 f32 VGPR layout: lanes 0-15 -> K={k0,k0+1}, lanes 16-31 -> K={k0+2,k0+3}
    const int row  = base + (lane & 15);
    const int coff = (lane >> 4) * 2;

    v8f c = {0.f, 0.f, 0.f, 0.f, 0.f, 0.f, 0.f, 0.f};

    const long long rb = (long long)row * NF + coff;
    accum_term(f0 + rb, w0 + coff, c);
    accum_term(f1 + rb, w1 + coff, c);
    accum_term(f2 + rb, w2 + coff, c);

    const float bias = b1p[0] + b2p[0];

    // C/D layout: lanes 0-15 hold rows 0-7 in c[0..7]; lanes 16-31 hold rows 8-15.
    // All 16 columns are identical (B was a broadcast), so pick m == sub.
    const int sub  = lane & 15;
    const int half = lane >> 4;
    if (sub < 8) {
        float dot = 0.0f;
#pragma unroll
        for (int m = 0; m < 8; ++m)
            if (sub == m) dot = c[m];           // compile-time extracts + cndmask

        const int atom = base + half * 8 + sub;
        const float lin = dot + bias;           // total_lin (== total_quad per ref bug)
        const float q   = charges[atom];
        const float tq  = lin * q;
        const float e   = tq * tq + lin;        // (total_quad*q)^2 + total_lin

        atom_out[atom] = e;
        atomicAdd(&mol_out[mol_index[atom]], e); // segment_sum (mol_index sorted, low contention)
    }
}

extern "C" void kernel_launch(void* const* d_in, const int* in_sizes, int n_in,
                              void* d_out, int out_size, void* d_ws, size_t ws_size,
                              hipStream_t stream) {
    const float* charges   = (const float*)d_in[0];
    const float* f0        = (const float*)d_in[1];
    const float* f1        = (const float*)d_in[2];
    const float* f2        = (const float*)d_in[3];
    const int*   mol_index = (const int*)  d_in[4];
    // d_in[5] = n_molecules (derived from out_size instead; no device read needed)
    const float* w0        = (const float*)d_in[6];
    const float* w1        = (const float*)d_in[7];
    const float* w2        = (const float*)d_in[8];
    const float* b1        = (const float*)d_in[9];
    const float* b2        = (const float*)d_in[10];

    const int n_atoms = in_sizes[0];
    const int n_mol   = out_size - n_atoms;

    float* mol_out  = (float*)d_out;        // [n_mol]
    float* atom_out = mol_out + n_mol;      // [n_atoms]

    // Zero the molecule accumulators (d_out is poisoned by the harness).
    {
        int threads = 256;
        int blocks = (n_mol + threads - 1) / threads;
        zero_f32_kernel<<<blocks, threads, 0, stream>>>(mol_out, n_mol);
    }

    // 16 atoms per wave, 8 waves (256 threads) per block.
    {
        int waves  = (n_atoms + 15) / 16;
        int blocks = (waves + 7) / 8;
        charge_energy_kernel<<<blocks, 256, 0, stream>>>(
            charges, f0, f1, f2, mol_index, w0, w1, w2, b1, b2,
            mol_out, atom_out, n_atoms);
    }
}